// CP_PINN_36850819400604
// MI455X (gfx1250) — compile-verified
//
#include <hip/hip_runtime.h>
#include <hip/hip_bf16.h>

typedef float v2f __attribute__((ext_vector_type(2)));
typedef float v8f __attribute__((ext_vector_type(8)));

#define NPTS 320
#define HID  128
#define RANK 64

// ---------------------------------------------------------------------------
// Kernel 1: per-axis MLP. One block per (dim, 16-point tile).
// f_out layout: (3, RANK, NPTS)  -> f[d][r][n]
// ---------------------------------------------------------------------------
__global__ __launch_bounds__(256) void cp_pinn_mlp_kernel(
    const float* __restrict__ xs,   // (3, 320, 1)
    const float* __restrict__ W0,   // (3, 1, 128)
    const float* __restrict__ b0,   // (3, 128)
    const float* __restrict__ W1,   // (3, 128, 128)
    const float* __restrict__ b1,   // (3, 128)
    const float* __restrict__ W2,   // (3, 128, 128)
    const float* __restrict__ b2,   // (3, 128)
    const float* __restrict__ W3,   // (3, 128, 64)
    const float* __restrict__ b3,   // (3, 64)
    float* __restrict__ f_out)      // (3, 64, 320)
{
    __shared__ float hA[16][HID];
    __shared__ float hB[16][HID];

    const int blk = blockIdx.x;          // 0..59
    const int d   = blk / 20;
    const int n0  = (blk % 20) * 16;

    const int t     = threadIdx.x;       // 0..255
    const int j     = t & (HID - 1);     // 0..127
    const int phalf = t >> 7;            // 0..1

    // ---- layer 0: hA[p][j] = tanh(x[p] * W0[j] + b0[j]) ----
    {
        const float* W0d = W0 + d * HID;
        const float* b0d = b0 + d * HID;
        const float  w   = W0d[j];
        const float  bb  = b0d[j];
        #pragma unroll
        for (int i = 0; i < 8; ++i) {
            const int p = phalf * 8 + i;
            const float x = xs[d * NPTS + n0 + p];
            hA[p][j] = tanhf(fmaf(x, w, bb));
        }
    }
    __syncthreads();

    // ---- layer 1: hB = tanh(hA @ W1 + b1) ----
    {
        const float* W1d = W1 + d * HID * HID;
        const float  bb  = b1[d * HID + j];
        #pragma unroll
        for (int i = 0; i < 8; ++i) {
            const int p = phalf * 8 + i;
            float acc = bb;
            #pragma unroll 8
            for (int k = 0; k < HID; ++k)
                acc = fmaf(hA[p][k], W1d[k * HID + j], acc);
            hB[p][j] = tanhf(acc);
        }
    }
    __syncthreads();

    // ---- layer 2: hA = tanh(hB @ W2 + b2) ----
    {
        const float* W2d = W2 + d * HID * HID;
        const float  bb  = b2[d * HID + j];
        #pragma unroll
        for (int i = 0; i < 8; ++i) {
            const int p = phalf * 8 + i;
            float acc = bb;
            #pragma unroll 8
            for (int k = 0; k < HID; ++k)
                acc = fmaf(hB[p][k], W2d[k * HID + j], acc);
            hA[p][j] = tanhf(acc);
        }
    }
    __syncthreads();

    // ---- layer 3: f[d][r][n0+p] = hA[p][:] @ W3[:, r] + b3[r] ----
    {
        const float* W3d = W3 + d * HID * RANK;
        const int r  = t & (RANK - 1);   // 0..63
        const int pq = t >> 6;           // 0..3
        const float bb = b3[d * RANK + r];
        #pragma unroll
        for (int i = 0; i < 4; ++i) {
            const int p = pq * 4 + i;    // 0..15
            float acc = bb;
            #pragma unroll 8
            for (int k = 0; k < HID; ++k)
                acc = fmaf(hA[p][k], W3d[k * RANK + r], acc);
            f_out[(d * RANK + r) * NPTS + n0 + p] = acc;
        }
    }
}

// ---------------------------------------------------------------------------
// Kernel 2: CP reconstruction via FP32 WMMA.
// out[a][b][c] = sum_r (f0[r][a]*f1[r][b]) * f2[r][c]
// Block = (a, 16-row b-tile). 4 waves; wave w covers c in [w*80, w*80+80).
// A (16x4 f32 frag): lane m=lane&15 is M-row; lanes<16 hold K {kk,kk+1},
// lanes>=16 hold K {kk+2,kk+3} in VGPR {0,1}. B is the symmetric transpose.
// ---------------------------------------------------------------------------
__global__ __launch_bounds__(128) void cp_pinn_wmma_kernel(
    const float* __restrict__ f,    // (3, 64, 320)
    float* __restrict__ out)        // (320, 320, 320)
{
    const float* f0 = f;
    const float* f1 = f + RANK * NPTS;
    const float* f2 = f + 2 * RANK * NPTS;

    const int bi = blockIdx.x;       // 0..6399
    const int a  = bi / 20;
    const int b0 = (bi % 20) * 16;

    __shared__ float As[RANK * 16];  // A panel: As[k*16 + m] = f0[k][a]*f1[k][b0+m]

    const int t = threadIdx.x;       // 0..127
    for (int e = t; e < RANK * 16; e += 128) {
        const int k = e >> 4;
        const int m = e & 15;
        As[e] = f0[k * NPTS + a] * f1[k * NPTS + b0 + m];
    }
    __syncthreads();

    const int lane = t & 31;
    const int wave = t >> 5;         // 0..3
    const int m    = lane & 15;      // M-row (A) / N-col (B)
    const int kh   = lane >> 4;      // K-half selector

    // Load all 16 A fragments (full K=64) into registers once; reuse 5x.
    v2f afrag[16];
    #pragma unroll
    for (int s = 0; s < 16; ++s) {
        const int kb = s * 4 + 2 * kh;
        afrag[s].x = As[kb * 16 + m];
        afrag[s].y = As[(kb + 1) * 16 + m];
    }

    float* outA = out + (size_t)a * NPTS * NPTS;

    #pragma unroll
    for (int ci = 0; ci < 5; ++ci) {
        const int c0 = (wave * 5 + ci) * 16;
        v8f acc = {0.f, 0.f, 0.f, 0.f, 0.f, 0.f, 0.f, 0.f};
        #pragma unroll
        for (int s = 0; s < 16; ++s) {
            const int kb = s * 4 + 2 * kh;
            v2f bfrag;
            bfrag.x = f2[kb * NPTS + c0 + m];        // L2-resident (80 KB)
            bfrag.y = f2[(kb + 1) * NPTS + c0 + m];
            // D = A x B + C, full fp32. 8 args:
            // (neg_a, A, neg_b, B, c_mod, C, reuse_a, reuse_b)
            acc = __builtin_amdgcn_wmma_f32_16x16x4_f32(
                false, afrag[s], false, bfrag, (short)0, acc, false, false);
        }
        // D layout: VGPR g -> row b0+g (lanes 0-15) / b0+g+8 (lanes 16-31),
        // column = c0 + (lane & 15).
        #pragma unroll
        for (int g = 0; g < 8; ++g) {
            const int row = b0 + g + kh * 8;
            outA[(size_t)row * NPTS + c0 + m] = acc[g];
        }
    }
}

// ---------------------------------------------------------------------------
extern "C" void kernel_launch(void* const* d_in, const int* in_sizes, int n_in,
                              void* d_out, int out_size, void* d_ws, size_t ws_size,
                              hipStream_t stream) {
    (void)in_sizes; (void)n_in; (void)out_size; (void)ws_size;

    const float* xs = (const float*)d_in[0];
    const float* W0 = (const float*)d_in[1];
    const float* b0 = (const float*)d_in[2];
    const float* W1 = (const float*)d_in[3];
    const float* b1 = (const float*)d_in[4];
    const float* W2 = (const float*)d_in[5];
    const float* b2 = (const float*)d_in[6];
    const float* W3 = (const float*)d_in[7];
    const float* b3 = (const float*)d_in[8];

    float* out = (float*)d_out;
    float* fws = (float*)d_ws;       // (3, 64, 320) = 245,760 bytes

    cp_pinn_mlp_kernel<<<3 * (NPTS / 16), 256, 0, stream>>>(
        xs, W0, b0, W1, b1, W2, b2, W3, b3, fws);

    cp_pinn_wmma_kernel<<<NPTS * (NPTS / 16), 128, 0, stream>>>(fws, out);
}